// MultiPf_32469952758284
// MI455X (gfx1250) — compile-verified
//
#include <hip/hip_runtime.h>
#include <math.h>

typedef __attribute__((ext_vector_type(2))) float v2f;
typedef __attribute__((ext_vector_type(8))) float v8f;

#define NDIM   256
#define LDA    260      // padded LDS row stride (floats): +4 kills half-wave bank conflicts
#define NPFS_  4
#define BATCH_ 128
#define NFM    512
#define TPB    1024

// One workgroup per (mode s, batch b) pair: gather the 256x256 skew submatrix
// into LDS (260KB, enabled by CDNA5's 320KB/WGP), run 128 Parlett-Reid pivot
// steps entirely in LDS; the rank-2 trailing update uses
// v_wmma_f32_16x16x4_f32 on 16x16 tiles (U=[v,-tau,0,0], W=[tau;v;0;0]).
__global__ __launch_bounds__(TPB) void pfaffian_kernel(
    const float* __restrict__ F, const int* __restrict__ idx,
    float* __restrict__ ws) {
  __shared__ float A[NDIM * LDA];
  __shared__ float s_tau[NDIM];
  __shared__ float s_v[NDIM];
  __shared__ int   ridx[NDIM];
  __shared__ int   s_kp;
  __shared__ float s_piv;

  const int blk = blockIdx.x;            // = s*BATCH + b
  const int s   = blk / BATCH_;
  const int b   = blk % BATCH_;
  const int tid = threadIdx.x;

  const float* Fs  = F + (size_t)s * NFM * NFM;
  const int*   row = idx + b * NDIM;

  if (tid < NDIM) ridx[tid] = row[tid];
  __syncthreads();

  // Gather + antisymmetrize: A[i][j] = 0.5*(F[ri][rj] - F[rj][ri])
  for (int e = tid; e < NDIM * NDIM; e += TPB) {
    int i = e >> 8, j = e & (NDIM - 1);
    int ri = ridx[i], rj = ridx[j];
    float f1 = Fs[ri * NFM + rj];
    float f2 = Fs[rj * NFM + ri];
    A[i * LDA + j] = 0.5f * (f1 - f2);
  }
  __syncthreads();

  float sign   = 1.0f;   // all threads track redundantly from LDS broadcasts
  float logabs = 0.0f;

  const int   lane  = tid & 31;
  // Force scalar (SGPR) wave id: makes the tile loop trip count uniform, so
  // the compiler uses scalar branches and EXEC stays all-1s at the WMMA.
  const int   wave  = __builtin_amdgcn_readfirstlane(tid >> 5);
  const int   l16   = lane & 15;
  const float maskf = (lane < 16) ? 1.0f : 0.0f;  // zero K=2,3 rows (lanes 16-31)
  const int   hoff  = (lane < 16) ? 0 : 8;        // C/D row offset for upper half-wave

  for (int kk = 0; kk < NDIM / 2; ++kk) {
    const int k = 2 * kk;

    // ---- phase 1 (wave 0): argmax_{i>k} |A[i][k]|; pivot = pre-swap A[k][kp]
    // (row swap never touches row k; col swap maps A[k][kp] -> A[k][k+1]).
    if (tid < 32) {
      float best  = -INFINITY;
      int   besti = 0x7fffffff;
      for (int i = (k & ~31) + tid; i < NDIM; i += 32) {
        if (i > k) {
          float a = fabsf(A[i * LDA + k]);
          if (a > best) { best = a; besti = i; }   // ascending i: keeps first max
        }
      }
      for (int off = 16; off > 0; off >>= 1) {
        float ob = __shfl_xor(best, off, 32);
        int   oi = __shfl_xor(besti, off, 32);
        if (ob > best || (ob == best && oi < besti)) { best = ob; besti = oi; }
      }
      if (tid == 0) {
        s_kp  = besti;
        s_piv = A[k * LDA + besti];
      }
    }
    __syncthreads();
    const int   kp  = s_kp;
    const float piv = s_piv;
    if (kp != k + 1) sign = -sign;
    sign   *= (piv > 0.0f) ? 1.0f : ((piv < 0.0f) ? -1.0f : 0.0f);
    logabs += logf(fabsf(piv));

    // ---- phase 2: row swap (k+1 <-> kp) ----
    if (tid < NDIM) {
      int c = tid;
      float t1 = A[(k + 1) * LDA + c];
      float t2 = A[kp * LDA + c];
      A[(k + 1) * LDA + c] = t2;
      A[kp * LDA + c]      = t1;
    }
    __syncthreads();

    // ---- phase 3: col swap fused with masked tau/v ----
    // Col swap only touches columns k+1,kp; tau reads column k (untouched);
    // v[r] is the post-swap A[r][k+1] == pre-swap A[r][kp] (same thread).
    if (tid < NDIM) {
      int r = tid;
      float t1 = A[r * LDA + (k + 1)];
      float t2 = A[r * LDA + kp];
      A[r * LDA + (k + 1)] = t2;
      A[r * LDA + kp]      = t1;
      bool m2 = r > (k + 1);
      s_tau[r] = m2 ? (A[r * LDA + k] / piv) : 0.0f;
      s_v[r]   = m2 ? t2 : 0.0f;
    }
    __syncthreads();

    // ---- phase 4: rank-2 trailing update via WMMA ----
    // dA = v(x)tau - tau(x)v  on tiles (ti,tj) in [t0,16)^2; masked tau/v make
    // partial tiles correct with no predication.
    const int t0    = (k + 2) >> 4;
    const int width = 16 - t0;
    const int count = width * width;
    for (int t = wave; t < count; t += 32) {   // scalar loop: EXEC all-1s
      int ti = t0 + t / width;
      int tj = t0 + t % width;
      int mrow = ti * 16 + l16;
      int ncol = tj * 16 + l16;
      // A operand (16x4): lanes 0-15 K=0,1 in .x,.y; lanes 16-31 K=2,3 = 0
      v2f a;
      a.x = maskf * s_v[mrow];
      a.y = -maskf * s_tau[mrow];
      // B operand (4x16): lanes 0-15 rows K=0,1; lanes 16-31 rows K=2,3 = 0
      v2f bmat;
      bmat.x = maskf * s_tau[ncol];
      bmat.y = maskf * s_v[ncol];
      // C/D (16x16 f32): vgpr r -> row M = r + 8*(lane>=16)
      v8f c;
      int rbase = ti * 16 + hoff;
      int cb    = tj * 16 + l16;
#pragma unroll
      for (int r = 0; r < 8; ++r) c[r] = A[(rbase + r) * LDA + cb];
      v8f d = __builtin_amdgcn_wmma_f32_16x16x4_f32(
          false, a, false, bmat, (short)0, c, false, false);
#pragma unroll
      for (int r = 0; r < 8; ++r) A[(rbase + r) * LDA + cb] = d[r];
    }
    __syncthreads();
  }

  if (tid == 0) {
    ws[blk * 2 + 0] = sign;
    ws[blk * 2 + 1] = logabs;
  }
}

// Combine over NPFS modes: signed logsumexp -> [sign(val); m+log|val|]
__global__ void combine_kernel(const float* __restrict__ ws,
                               float* __restrict__ out) {
  int b = threadIdx.x;
  if (b < BATCH_) {
    float sg[NPFS_], lg[NPFS_];
    float m = -INFINITY;
    for (int s = 0; s < NPFS_; ++s) {
      sg[s] = ws[(s * BATCH_ + b) * 2 + 0];
      lg[s] = ws[(s * BATCH_ + b) * 2 + 1];
      m = fmaxf(m, lg[s]);
    }
    float val = 0.0f;
    for (int s = 0; s < NPFS_; ++s) val += sg[s] * expf(lg[s] - m);
    out[b]          = (val > 0.0f) ? 1.0f : ((val < 0.0f) ? -1.0f : 0.0f);
    out[BATCH_ + b] = m + logf(fabsf(val));
  }
}

extern "C" void kernel_launch(void* const* d_in, const int* in_sizes, int n_in,
                              void* d_out, int out_size, void* d_ws, size_t ws_size,
                              hipStream_t stream) {
  (void)in_sizes; (void)n_in; (void)out_size; (void)ws_size;
  const float* F   = (const float*)d_in[0];   // [4, 512, 512] f32
  const int*   idx = (const int*)d_in[1];     // [128, 256] i32
  float* ws  = (float*)d_ws;                  // 512 * 2 floats
  float* out = (float*)d_out;                 // [2, 128] f32

  pfaffian_kernel<<<NPFS_ * BATCH_, TPB, 0, stream>>>(F, idx, ws);
  combine_kernel<<<1, BATCH_, 0, stream>>>(ws, out);
}